// KernelPredictConv2D_80814104641711
// MI455X (gfx1250) — compile-verified
//
#include <hip/hip_runtime.h>

typedef float v2f __attribute__((ext_vector_type(2)));
typedef float v8f __attribute__((ext_vector_type(8)));

#define B_    4
#define H_    256
#define W_    256
#define C_    32
#define K_    5
#define TAPS  25
#define TILE  16
#define WPB   8               // waves per block
#define BQ    20              // band width: TILE + K_ - 1
#define BROW  (TILE * BQ)     // 320 floats per di slab

// Inner accumulation over the 5 feature rows. WCHK=true adds W-boundary
// clamp+select (only the first/last tile of each row); interior tiles take
// the bare-load path. All loads are unconditional (no EXEC masking), so the
// WMMAs always execute with EXEC all-1s and the loop stays straight-line.
template <bool WCHK>
__device__ __forceinline__ void accum_rows(
    const float* __restrict__ feat, const float* aw,
    int b, int h, int w0, int idx, int k0,
    v8f& acc0, v8f& acc1)
{
#pragma unroll
    for (int di = 0; di < K_; ++di) {
        const int hh = h + di - 2;
        if (hh < 0 || hh >= H_) continue;               // uniform per wave
        const float* frow  = feat + (size_t)(b * H_ + hh) * (W_ * C_);
        const float* abase = aw + di * BROW + idx * BQ;

#pragma unroll
        for (int kc = 0; kc < K_; ++kc) {
            const int q0 = kc * 4 + k0;                  // even -> 8B aligned
            const v2f a  = *reinterpret_cast<const v2f*>(abase + q0); // ds_load_b64

            const int wq = w0 - 2 + q0;
            v2f b0, b1;
            if (WCHK) {
                const int wc0 = min(max(wq, 0), W_ - 1);
                const int wc1 = min(max(wq + 1, 0), W_ - 1);
                const bool v0 = (wq == wc0);
                const bool v1 = (wq + 1 == wc1);
                const float f00 = frow[wc0 * C_ + idx];
                const float f01 = frow[wc1 * C_ + idx];
                const float f10 = frow[wc0 * C_ + idx + 16];
                const float f11 = frow[wc1 * C_ + idx + 16];
                b0.x = v0 ? f00 : 0.0f;  b0.y = v1 ? f01 : 0.0f;
                b1.x = v0 ? f10 : 0.0f;  b1.y = v1 ? f11 : 0.0f;
            } else {
                b0.x = frow[wq * C_ + idx];
                b0.y = frow[(wq + 1) * C_ + idx];
                b1.x = frow[wq * C_ + idx + 16];
                b1.y = frow[(wq + 1) * C_ + idx + 16];
            }

            acc0 = __builtin_amdgcn_wmma_f32_16x16x4_f32(
                       false, a, false, b0, (short)0, acc0, false, false);
            acc1 = __builtin_amdgcn_wmma_f32_16x16x4_f32(
                       false, a, false, b1, (short)0, acc1, false, false);
        }
    }
}

__global__ __launch_bounds__(256) void KernelPredictConv2D_wmma(
    const float* __restrict__ feat,   // [B,H,W,C]
    const float* __restrict__ kern,   // [B,H,W,25]
    const float* __restrict__ bias,   // [C]
    float* __restrict__ out)          // [B,H,W,C]
{
    __shared__ float als[WPB][K_ * BROW];   // banded A, 6.25KB per wave

    const int lane = threadIdx.x & 31;
    const int wid  = threadIdx.x >> 5;
    const int half = lane >> 4;       // 0: lanes 0-15, 1: lanes 16-31
    const int idx  = lane & 15;       // M row (A) / N col (B,C,D)
    const int k0   = half * 2;        // K columns {k0, k0+1} held by this lane

    // tile id -> (b, h, w0); 8 consecutive tiles per block share (b,h).
    const int tile = blockIdx.x * WPB + wid;
    const int wt = tile & 15;
    const int h  = (tile >> 4) & (H_ - 1);
    const int b  = tile >> 12;
    const int w0 = wt * TILE;

    // Build zero-padded banded A in LDS (straight-line, clamp + select):
    //   als[di][m][q] = (0 <= q-m < 5) ? kern[b,h,w0+m, di*5 + (q-m)] : 0
    // Lane covers m = idx, q = {half, half+2, ..., half+18}.
    const float* kptr = kern + (size_t)((b * H_ + h) * W_ + w0) * TAPS;
    float* aw = &als[wid][0];
    {
        const int m = idx;
#pragma unroll
        for (int di = 0; di < K_; ++di) {
#pragma unroll
            for (int t = 0; t < BQ / 2; ++t) {
                const int q   = 2 * t + half;
                const int dj  = q - m;
                const int djc = min(max(dj, 0), K_ - 1);        // clamped, in-bounds
                float v = kptr[m * TAPS + di * K_ + djc];
                v = (dj == djc) ? v : 0.0f;
                aw[di * BROW + m * BQ + q] = v;
            }
        }
    }
    __syncthreads();

    v8f acc0 = {};   // channels 0-15
    v8f acc1 = {};   // channels 16-31

    if (w0 != 0 && w0 != W_ - TILE)
        accum_rows<false>(feat, aw, b, h, w0, idx, k0, acc0, acc1);
    else
        accum_rows<true>(feat, aw, b, h, w0, idx, k0, acc0, acc1);

    // Epilogue: bias add + store. C/D layout: VGPR r -> M = r + half*8, N = idx.
    const float bs0 = bias[idx];
    const float bs1 = bias[idx + 16];
    float* orow = out + (size_t)((b * H_ + h) * W_ + w0) * C_;
#pragma unroll
    for (int r = 0; r < 8; ++r) {
        const int m = r + half * 8;
        orow[(size_t)m * C_ + idx     ] = acc0[r] + bs0;
        orow[(size_t)m * C_ + idx + 16] = acc1[r] + bs1;
    }
}

extern "C" void kernel_launch(void* const* d_in, const int* in_sizes, int n_in,
                              void* d_out, int out_size, void* d_ws, size_t ws_size,
                              hipStream_t stream) {
    (void)in_sizes; (void)n_in; (void)out_size; (void)d_ws; (void)ws_size;
    const float* feat = (const float*)d_in[0];
    const float* kern = (const float*)d_in[1];
    const float* bias = (const float*)d_in[2];
    float* out = (float*)d_out;

    const int tiles  = B_ * H_ * (W_ / TILE);   // 16384
    const int blocks = tiles / WPB;             // 2048 blocks of 256 threads
    KernelPredictConv2D_wmma<<<blocks, 256, 0, stream>>>(feat, kern, bias, out);
}